// GRUByHandBidirectional_60687887892609
// MI455X (gfx1250) — compile-verified
//
#include <hip/hip_runtime.h>

typedef float v2f __attribute__((ext_vector_type(2)));
typedef float v8f __attribute__((ext_vector_type(8)));

#define HID 128
#define INF 512
#define NWG 8              // workgroup g owns hidden units [16g, 16g+16)
#define WAVES 16           // K-split across waves within a workgroup
#define TPB (WAVES * 32)
#define KX (INF / WAVES)   // 32-float x-GEMV K-slice per wave
#define KH (HID / WAVES)   // 8-float  h-GEMV K-slice per wave

// GRU cell, batch=1: gates via V_WMMA_F32_16X16X4_F32 (M=1 lives in row 0 of A).
// Per workgroup: 3 row tiles (r/z/n gate rows j0, j0+128, j0+256) x 2 GEMVs,
// K split over 16 waves, partials reduced in LDS, gate nonlinearity fused.
__global__ __launch_bounds__(TPB) void gru_cell_wmma(
    const float* __restrict__ x_t,   // (1,512)
    const float* __restrict__ h0,    // (2,1,128); forward dir = first 128
    const float* __restrict__ w_ih,  // (384,512) row-major
    const float* __restrict__ w_hh,  // (384,128) row-major
    const float* __restrict__ b_ih,  // (384,)
    const float* __restrict__ b_hh,  // (384,)
    float* __restrict__ out)         // (1,128)
{
    __shared__ float xs[INF];
    __shared__ float hs[HID];
    __shared__ float zbuf[KX + 2];        // zero source: lanes n!=0 stride thru this
    __shared__ float part[6][WAVES][16];  // 3 x-tiles + 3 h-tiles, per-wave partials
    __shared__ float red[6][16];

    const int tid = threadIdx.x;
    const int g   = blockIdx.x;

    // Stage activations + zero pad into LDS.
    if (tid < INF)      xs[tid]   = x_t[tid];
    if (tid < HID)      hs[tid]   = h0[tid];     // h0[0] = forward hidden state
    if (tid < KX + 2)   zbuf[tid] = 0.0f;
    __syncthreads();

    const int wave = tid >> 5;
    const int lane = tid & 31;
    const int n    = lane & 15;   // N index (weight row within 16-row tile)
    const int hi   = lane >> 4;   // half-wave: K pair {0,1} vs {2,3}

    // Row tiles this workgroup owns: r, z, n gate rows for hidden block 16g.
    const int jbase0 = 16 * g;

    #pragma unroll
    for (int t = 0; t < 3; ++t) {
        const int j0 = t * HID + jbase0;

        // ---- x-GEMV K-slice: 8 WMMAs, fully unrolled ----
        {
            v8f acc = {};
            const float* bp = w_ih + (size_t)(j0 + n) * INF + wave * KX + 2 * hi;
            const float* ap = (n == 0) ? &xs[wave * KX + 2 * hi] : &zbuf[2 * hi];
            #pragma unroll
            for (int kb = 0; kb < KX; kb += 4) {
                v2f b = *(const v2f*)(bp + kb);   // global_load_b64, imm offsets
                v2f a = *(const v2f*)(ap + kb);   // ds_load_b64, imm offsets
                acc = __builtin_amdgcn_wmma_f32_16x16x4_f32(
                    false, a, false, b, (short)0, acc, false, false);
            }
            if (lane < 16) part[t][wave][lane] = acc[0];  // D row M=0
        }

        // ---- h-GEMV K-slice: 2 WMMAs ----
        {
            v8f acc = {};
            const float* bp = w_hh + (size_t)(j0 + n) * HID + wave * KH + 2 * hi;
            const float* ap = (n == 0) ? &hs[wave * KH + 2 * hi] : &zbuf[2 * hi];
            #pragma unroll
            for (int kb = 0; kb < KH; kb += 4) {
                v2f b = *(const v2f*)(bp + kb);
                v2f a = *(const v2f*)(ap + kb);
                acc = __builtin_amdgcn_wmma_f32_16x16x4_f32(
                    false, a, false, b, (short)0, acc, false, false);
            }
            if (lane < 16) part[3 + t][wave][lane] = acc[0];
        }
    }
    __syncthreads();

    // Reduce 16 wave-partials for each of the 6 (array, col) groups.
    if (tid < 96) {
        const int a = tid >> 4, c = tid & 15;
        float s = 0.0f;
        #pragma unroll
        for (int w2 = 0; w2 < WAVES; ++w2) s += part[a][w2][c];
        red[a][c] = s;
    }
    __syncthreads();

    // Fused gate combine for this workgroup's 16 hidden units.
    if (tid < 16) {
        const int jr = jbase0 + tid;
        const int jz = HID + jbase0 + tid;
        const int jn = 2 * HID + jbase0 + tid;
        const float rx = red[0][tid] + b_ih[jr], rh = red[3][tid] + b_hh[jr];
        const float zx = red[1][tid] + b_ih[jz], zh = red[4][tid] + b_hh[jz];
        const float nx = red[2][tid] + b_ih[jn], nh = red[5][tid] + b_hh[jn];
        const float r  = 1.0f / (1.0f + __expf(-(rx + rh)));
        const float z  = 1.0f / (1.0f + __expf(-(zx + zh)));
        const float nn = tanhf(nx + r * nh);
        out[jbase0 + tid] = (1.0f - z) * nn + z * hs[jbase0 + tid];
    }
}

extern "C" void kernel_launch(void* const* d_in, const int* in_sizes, int n_in,
                              void* d_out, int out_size, void* d_ws, size_t ws_size,
                              hipStream_t stream) {
    const float* x_t  = (const float*)d_in[0];
    const float* h0   = (const float*)d_in[1];
    const float* w_ih = (const float*)d_in[2];
    const float* w_hh = (const float*)d_in[3];
    const float* b_ih = (const float*)d_in[4];
    const float* b_hh = (const float*)d_in[5];
    float* out = (float*)d_out;

    gru_cell_wmma<<<NWG, TPB, 0, stream>>>(x_t, h0, w_ih, w_hh, b_ih, b_hh, out);
}